// Cut_Blance_loss_39814346834349
// MI455X (gfx1250) — compile-verified
//
#include <hip/hip_runtime.h>

typedef float v2f __attribute__((ext_vector_type(2)));
typedef float v8f __attribute__((ext_vector_type(8)));
typedef int   v4i __attribute__((vector_size(16)));

#define N_TOT   16384
#define G_TOT   16
#define KC      64            // K chunk per stage
#define ROWS_WG 64            // 4 waves * 16 rows
#define TPB     128
#define LSTRIDE 68            // A-tile LDS row stride (dwords): mult of 4, conflict-free frags
#define NCHUNK  (N_TOT / KC)  // 256
#define ABUF    (ROWS_WG * LSTRIDE)   // floats per A stage buffer
#define YBUF    (KC * G_TOT)          // floats per Y stage buffer (row-major [k][g])

#if defined(__has_builtin) && __has_builtin(__builtin_amdgcn_global_load_async_to_lds_b128)
#define HAVE_ASYNC_LDS 1
#else
#define HAVE_ASYNC_LDS 0
#endif

template <int N>
__device__ __forceinline__ void wait_asynccnt() {
#if defined(__has_builtin) && __has_builtin(__builtin_amdgcn_s_wait_asynccnt)
    __builtin_amdgcn_s_wait_asynccnt(N);
#else
    asm volatile("s_wait_asynccnt %0" :: "i"(N) : "memory");
#endif
}

// ---------------------------------------------------------------------------
// Kernel 0: zero the 48-float accumulator workspace (numer[16], gamma[16], ysum[16])
// ---------------------------------------------------------------------------
__global__ void cutbal_zero(float* ws) {
    if (threadIdx.x < 48) ws[threadIdx.x] = 0.0f;
}

// ---------------------------------------------------------------------------
// Kernel 1: fused streaming pass over A.
//  - per wave: 16 rows of A, full K sweep
//  - WMMA f32 16x16x4 accumulates AY = A@Y tile (16 rows x 16 groups)
//  - D (row sums of A) folded in with one packed v_pk_add_f32 per fragment
//  - ALL staging is async global->LDS (ASYNCcnt), triple-buffered, 2 chunks ahead
//  - epilogue peeled so the hot loop has an unconditional wait (no branches)
//  - tail: numer/Gamma/colsum partials -> global fp32 atomics
// ---------------------------------------------------------------------------
__global__ __launch_bounds__(TPB)
void cutbal_main(const float* __restrict__ Yg, const float* __restrict__ Ag,
                 float* __restrict__ acc)
{
    __shared__ __align__(16) float lds_a[3 * ABUF];   // A tiles, padded rows
    __shared__ __align__(16) float lds_y[3 * YBUF];   // Y chunks, row-major [k][g]

    const int t    = threadIdx.x;
    const int lane = t & 31;
    const int wv   = t >> 5;        // wave 0..3
    const int l15  = lane & 15;
    const int hi   = lane >> 4;     // half-wave select (K pair)
    const int r0   = blockIdx.x * ROWS_WG;

    // staging assignment: thread t stages 32 consecutive floats of one A row half,
    // plus 8 consecutive floats of the Y chunk. 10 async b128 issues per chunk.
    const int arow = t >> 1;
    const int acol = (t & 1) * 32;
    const float* agsrc = Ag + (size_t)(r0 + arow) * N_TOT + acol;
    const float* ygsrc = Yg + t * 8;

    v8f c     = {};        // 16x16 f32 accumulator (AY tile)
    v2f dacc2 = {};        // packed partial row sum of A (this lane's K subset)

    // one chunk of WMMA work from staged buffer `p`
    auto compute_chunk = [&](int p) {
        const float* pa = &lds_a[p * ABUF + (wv * 16 + l15) * LSTRIDE + 2 * hi];
        const float* pb = &lds_y[p * YBUF + (hi << 5) + l15];   // row-major: K=2*hi, N=l15
        #pragma unroll
        for (int j = 0; j < KC / 4; ++j) {
            v2f a = *(const v2f*)(pa + 4 * j);
            v2f b = { pb[j * 64], pb[j * 64 + 16] };    // K = 4j+2hi, 4j+2hi+1
            dacc2 = dacc2 + a;                          // packed row-sum fold
            c = __builtin_amdgcn_wmma_f32_16x16x4_f32(
                    false, a, false, b, (short)0, c, false, false);
        }
    };

#if HAVE_ASYNC_LDS
    #define STAGE_CHUNK(K0, BUF)                                                        \
        do {                                                                            \
            v4i* gs = (v4i*)const_cast<float*>(agsrc + (K0));                           \
            v4i* ls = (v4i*)&lds_a[(BUF) * ABUF + arow * LSTRIDE + acol];               \
            __builtin_amdgcn_global_load_async_to_lds_b128(gs, ls,   0, 0);             \
            __builtin_amdgcn_global_load_async_to_lds_b128(gs, ls,  16, 0);             \
            __builtin_amdgcn_global_load_async_to_lds_b128(gs, ls,  32, 0);             \
            __builtin_amdgcn_global_load_async_to_lds_b128(gs, ls,  48, 0);             \
            __builtin_amdgcn_global_load_async_to_lds_b128(gs, ls,  64, 0);             \
            __builtin_amdgcn_global_load_async_to_lds_b128(gs, ls,  80, 0);             \
            __builtin_amdgcn_global_load_async_to_lds_b128(gs, ls,  96, 0);             \
            __builtin_amdgcn_global_load_async_to_lds_b128(gs, ls, 112, 0);             \
            v4i* gy = (v4i*)const_cast<float*>(ygsrc + (size_t)(K0) * G_TOT);           \
            v4i* ly = (v4i*)&lds_y[(BUF) * YBUF + t * 8];                               \
            __builtin_amdgcn_global_load_async_to_lds_b128(gy, ly,   0, 0);             \
            __builtin_amdgcn_global_load_async_to_lds_b128(gy, ly,  16, 0);             \
        } while (0)

    // prologue: prefetch chunks 0 and 1
    STAGE_CHUNK(0, 0);
    STAGE_CHUNK(KC, 1);
    wait_asynccnt<10>();        // chunk 0 resident; chunk 1's 10 still in flight
    __syncthreads();

    // main loop: branch-free steady state, always 2 chunks ahead
    int p = 0;                  // buffer holding current chunk
    for (int i = 0; i + 2 < NCHUNK; ++i) {
        int q = p + 2; if (q >= 3) q -= 3;
        STAGE_CHUNK((i + 2) * KC, q);                   // prefetch 2 ahead
        compute_chunk(p);
        wait_asynccnt<10>();                            // chunk i+1 resident, i+2 in flight
        __syncthreads();
        ++p; if (p >= 3) p -= 3;
    }

    // peeled epilogue: last two chunks, no more staging
    compute_chunk(p);           // chunk NCHUNK-2 (resident per last wait<10>)
    wait_asynccnt<0>();         // drain chunk NCHUNK-1
    __syncthreads();
    ++p; if (p >= 3) p -= 3;
    compute_chunk(p);           // chunk NCHUNK-1 (tail reads no LDS -> no final barrier)
    #undef STAGE_CHUNK
#else
    // fallback: synchronous single-buffer staging (VGPR roundtrip)
    for (int k0 = 0; k0 < N_TOT; k0 += KC) {
        #pragma unroll
        for (int u = 0; u < 8; ++u) {
            float4 v = *(const float4*)(agsrc + k0 + 4 * u);
            *(float4*)&lds_a[arow * LSTRIDE + acol + 4 * u] = v;
        }
        #pragma unroll
        for (int u = 0; u < 2; ++u) {
            float4 v = *(const float4*)(ygsrc + (size_t)k0 * G_TOT + 4 * u);
            *(float4*)&lds_y[t * 8 + 4 * u] = v;
        }
        __syncthreads();
        compute_chunk(0);
        __syncthreads();
    }
#endif

    // ---- tail: per-wave reductions ----
    // complete D for rows l15 (horizontal add + combine the two K half-subsets)
    float dh    = dacc2.x + dacc2.y;
    float dfull = dh + __shfl_xor(dh, 16);

    const int rowbase = r0 + wv * 16;
    float np = 0.0f, gp = 0.0f, yp = 0.0f;   // numer / Gamma / colsum partials, group = l15
    #pragma unroll
    for (int r = 0; r < 8; ++r) {
        int   m  = r + (hi << 3);                        // output row held in c[r] on this lane
        float Dm = __shfl(dfull, m);                     // D[rowbase+m] (lanes 0..15 hold rows 0..15)
        float yv = Yg[(size_t)(rowbase + m) * G_TOT + l15];
        np += yv * (Dm - c[r]);                          // Y * (A@(1-Y)) = Y*(D - A@Y)
        gp += yv * Dm;                                   // Y^T D
        yp += yv;                                        // colsum(Y)
    }
    unsafeAtomicAdd(&acc[l15],      np);
    unsafeAtomicAdd(&acc[16 + l15], gp);
    unsafeAtomicAdd(&acc[32 + l15], yp);
}

// ---------------------------------------------------------------------------
// Kernel 2: finalize scalar loss
// ---------------------------------------------------------------------------
__global__ void cutbal_finalize(const float* __restrict__ ws, float* __restrict__ out) {
    if (threadIdx.x == 0) {
        float cut = 0.0f, bal = 0.0f;
        const float target = (float)N_TOT / (float)G_TOT;   // 1024
        for (int i = 0; i < 16; ++i) {
            cut += ws[i] / ws[16 + i];
            float d = ws[32 + i] - target;
            bal += d * d;
        }
        out[0] = cut + bal;
    }
}

// ---------------------------------------------------------------------------
extern "C" void kernel_launch(void* const* d_in, const int* in_sizes, int n_in,
                              void* d_out, int out_size, void* d_ws, size_t ws_size,
                              hipStream_t stream) {
    const float* Y = (const float*)d_in[0];   // [16384, 16] f32
    const float* A = (const float*)d_in[1];   // [16384, 16384] f32
    float* acc = (float*)d_ws;                // 48 floats
    float* out = (float*)d_out;               // scalar f32

    cutbal_zero<<<1, 64, 0, stream>>>(acc);
    cutbal_main<<<N_TOT / ROWS_WG, TPB, 0, stream>>>(Y, A, acc);
    cutbal_finalize<<<1, 32, 0, stream>>>(acc, out);
}